// HeterogeneousAggregationLayers_42099269435817
// MI455X (gfx1250) — compile-verified
//
#include <hip/hip_runtime.h>

typedef float v2f __attribute__((ext_vector_type(2)));
typedef float v8f __attribute__((ext_vector_type(8)));

#define FEAT 64

// ---------------- utility kernels ----------------

__global__ void fill_zero_kernel(float* __restrict__ p, int n) {
  int i = blockIdx.x * blockDim.x + threadIdx.x;
  if (i < n) p[i] = 0.0f;
}

__global__ void copy_kernel(const float4* __restrict__ in, float4* __restrict__ out, int n4) {
  int i = blockIdx.x * blockDim.x + threadIdx.x;
  if (i < n4) out[i] = in[i];
}

__global__ void degree_kernel(const int* __restrict__ src, const int* __restrict__ dst,
                              float* __restrict__ degE, float* __restrict__ degU, int nEd) {
  int i = blockIdx.x * blockDim.x + threadIdx.x;
  if (i < nEd) {
    unsafeAtomicAdd(&degE[src[i]], 1.0f);
    unsafeAtomicAdd(&degU[dst[i]], 1.0f);
  }
}

// out[idx] /= (deg[row] + 1)
__global__ void normalize_kernel(float* __restrict__ x, const float* __restrict__ deg, int n) {
  int i = blockIdx.x * blockDim.x + threadIdx.x;
  if (i < n) x[i] *= __frcp_rn(deg[i >> 6] + 1.0f);
}

// ---------------- WMMA linear: Y = X @ W^T + b ----------------
// One wave computes a 16-row tile of Y (all 64 output columns, 4 N-tiles).
// V_WMMA_F32_16X16X4_F32: A 16x4, B 4x16, C/D 16x16 f32.
// A frag (ISA 7.12.2, 32-bit A 16x4): lane l (l<16): row M=l, holds K=0,1;
// lanes 16..31: row M=l-16, holds K=2,3.  B mirrors A (row striped across lanes).
__global__ void linear_wmma_kernel(const float* __restrict__ X, const float* __restrict__ W,
                                   const float* __restrict__ bias, float* __restrict__ Y,
                                   int nrows) {
  const int lane = threadIdx.x & 31;
  const int wave = (blockIdx.x * blockDim.x + threadIdx.x) >> 5;
  const int m0 = wave * 16;
  if (m0 >= nrows) return;  // wave-uniform: EXEC stays all-1s for WMMA

  const int half = lane >> 4;   // which K-pair this lane holds
  const int l16 = lane & 15;

  // Preload A fragments for all 16 K-steps: X[m0+l16, 4*ks + 2*half .. +1]
  v2f afrag[16];
  const float* xrow = X + (size_t)(m0 + l16) * FEAT + 2 * half;
#pragma unroll
  for (int ks = 0; ks < 16; ++ks) afrag[ks] = *(const v2f*)(xrow + ks * 4);

#pragma unroll
  for (int nt = 0; nt < 4; ++nt) {
    const int n0 = nt * 16;
    // C init = bias[n0 + N]; each lane's 8 acc elements share column N = l16
    const float bv = bias[n0 + l16];
    v8f c = {bv, bv, bv, bv, bv, bv, bv, bv};
    // B[k, n] = W^T[k, n] = W[n0+n, k]: lane n loads W row (n0+n), k = 4*ks + 2*half
    const float* wrow = W + (size_t)(n0 + l16) * FEAT + 2 * half;
#pragma unroll
    for (int ks = 0; ks < 16; ++ks) {
      v2f bfrag = *(const v2f*)(wrow + ks * 4);
      c = __builtin_amdgcn_wmma_f32_16x16x4_f32(false, afrag[ks], false, bfrag,
                                                (short)0, c, false, false);
    }
    // D layout: lane holds column N=l16; VGPR v -> row m0 + v + 8*half
    float* ybase = Y + (size_t)(m0 + 8 * half) * FEAT + n0 + l16;
#pragma unroll
    for (int v = 0; v < 8; ++v) ybase[(size_t)v * FEAT] = c[v];
  }
}

// ---------------- edge scatter ----------------
// thread idx -> (edge e, feature c). Adds Ae[src]->accU[dst], Au[dst]->accE[src].
__global__ void scatter_kernel(const int* __restrict__ src, const int* __restrict__ dst,
                               const float* __restrict__ Au, const float* __restrict__ Ae,
                               float* __restrict__ accU, float* __restrict__ accE, int nEd) {
  int idx = blockIdx.x * blockDim.x + threadIdx.x;
  if (idx < nEd * FEAT) {
    const int e = idx >> 6;
    const int c = idx & 63;
    const int s = src[e];
    const int d = dst[e];
    unsafeAtomicAdd(&accU[(size_t)d * FEAT + c], Ae[(size_t)s * FEAT + c]);
    unsafeAtomicAdd(&accE[(size_t)s * FEAT + c], Au[(size_t)d * FEAT + c]);
  }
}

// ---------------- launcher ----------------

extern "C" void kernel_launch(void* const* d_in, const int* in_sizes, int n_in,
                              void* d_out, int out_size, void* d_ws, size_t ws_size,
                              hipStream_t stream) {
  const float* x_user  = (const float*)d_in[0];
  const float* x_event = (const float*)d_in[1];
  const int*   src     = (const int*)d_in[2];
  const int*   dst     = (const int*)d_in[3];
  const float* Wu0 = (const float*)d_in[4];
  const float* bu0 = (const float*)d_in[5];
  const float* We0 = (const float*)d_in[6];
  const float* be0 = (const float*)d_in[7];
  const float* Wu1 = (const float*)d_in[8];
  const float* bu1 = (const float*)d_in[9];
  const float* We1 = (const float*)d_in[10];
  const float* be1 = (const float*)d_in[11];

  const int nU  = in_sizes[0] / FEAT;   // 200000
  const int nEv = in_sizes[1] / FEAT;   // 50000
  const int nEd = in_sizes[2];          // 1000000

  // workspace: linear outputs + degree arrays (accumulators live in d_out)
  float* Au   = (float*)d_ws;                  // nU  * 64
  float* Ae   = Au + (size_t)nU * FEAT;        // nEv * 64
  float* degU = Ae + (size_t)nEv * FEAT;       // nU
  float* degE = degU + nU;                     // nEv (contiguous after degU)

  float* outU = (float*)d_out;
  float* outE = outU + (size_t)nU * FEAT;

  const int B = 256;

  // degrees (same for both layers)
  {
    int n = nU + nEv;
    fill_zero_kernel<<<(n + B - 1) / B, B, 0, stream>>>(degU, n);
    degree_kernel<<<(nEd + B - 1) / B, B, 0, stream>>>(src, dst, degE, degU, nEd);
  }

  for (int layer = 0; layer < 2; ++layer) {
    const float* inU = (layer == 0) ? x_user  : outU;
    const float* inE = (layer == 0) ? x_event : outE;
    const float* Wu  = (layer == 0) ? Wu0 : Wu1;
    const float* bu  = (layer == 0) ? bu0 : bu1;
    const float* We  = (layer == 0) ? We0 : We1;
    const float* be  = (layer == 0) ? be0 : be1;

    // linears (WMMA)
    {
      int waves = (nU + 15) / 16;
      linear_wmma_kernel<<<(waves * 32 + B - 1) / B, B, 0, stream>>>(inU, Wu, bu, Au, nU);
    }
    {
      int waves = (nEv + 15) / 16;
      linear_wmma_kernel<<<(waves * 32 + B - 1) / B, B, 0, stream>>>(inE, We, be, Ae, nEv);
    }
    // residual init: acc = linear output
    {
      int n4 = nU * (FEAT / 4);
      copy_kernel<<<(n4 + B - 1) / B, B, 0, stream>>>((const float4*)Au, (float4*)outU, n4);
    }
    {
      int n4 = nEv * (FEAT / 4);
      copy_kernel<<<(n4 + B - 1) / B, B, 0, stream>>>((const float4*)Ae, (float4*)outE, n4);
    }
    // edge scatter-add (both directions)
    {
      int n = nEd * FEAT;  // 64M < 2^31
      scatter_kernel<<<(n + B - 1) / B, B, 0, stream>>>(src, dst, Au, Ae, outU, outE, nEd);
    }
    // normalize by degree+1
    {
      int n = nU * FEAT;
      normalize_kernel<<<(n + B - 1) / B, B, 0, stream>>>(outU, degU, n);
    }
    {
      int n = nEv * FEAT;
      normalize_kernel<<<(n + B - 1) / B, B, 0, stream>>>(outE, degE, n);
    }
  }
}